// JTMPN_24232205484224
// MI455X (gfx1250) — compile-verified
//
#include <hip/hip_runtime.h>
#include <hip/hip_bf16.h>

typedef float v2f __attribute__((ext_vector_type(2)));
typedef float v8f __attribute__((ext_vector_type(8)));

#define H      256
#define NBR    10
#define AFD    35
#define BFD    5
#define KIN    (AFD + BFD)   // 40
#define KOUT   (AFD + H)     // 291
#define NA     100000
#define NB_    200000
#define NM     20000
#define NMOLS  2000

// Each wave computes a 16x64 output tile: 4 independent accumulator chains so
// v_wmma ops pipeline back-to-back (no accumulator RAW between consecutive
// wmmas), and each A fragment (one b64 load) feeds 4 wmmas.

// ---------------------------------------------------------------------------
// GEMM1: binput = fbonds @ W_i ; gmsg = relu(binput)
// ---------------------------------------------------------------------------
__global__ __launch_bounds__(256) void k_gemm_in(const float* __restrict__ fb,
                                                 const float* __restrict__ Wi,
                                                 float* __restrict__ binput,
                                                 float* __restrict__ gmsg)
{
    const int lane  = threadIdx.x & 31;
    const int wave  = blockIdx.x * (blockDim.x >> 5) + (threadIdx.x >> 5);
    const int tileM = wave >> 2;            // 4 col-groups of 64
    const int colBase = (wave & 3) * 64;
    if (tileM * 16 >= NB_) return;          // wave-uniform
    const int hs  = lane >> 4;
    const int l16 = lane & 15;
    const int aRow = tileM * 16 + l16;

    v8f c[4] = {};
    for (int k = 0; k < KIN; k += 4) {
        const int ka = k + hs * 2;
        v2f a;
        a.x = fb[(size_t)aRow * KIN + ka];
        a.y = fb[(size_t)aRow * KIN + ka + 1];
#pragma unroll
        for (int j = 0; j < 4; ++j) {
            const int col = colBase + j * 16 + l16;
            v2f b;
            b.x = Wi[(size_t)ka * H + col];
            b.y = Wi[(size_t)(ka + 1) * H + col];
            c[j] = __builtin_amdgcn_wmma_f32_16x16x4_f32(false, a, false, b,
                                                         (short)0, c[j], false, false);
        }
    }
#pragma unroll
    for (int j = 0; j < 4; ++j) {
        const int col = colBase + j * 16 + l16;
#pragma unroll
        for (int r = 0; r < 8; ++r) {
            const int row = tileM * 16 + r + hs * 8;
            const size_t off = (size_t)row * H + col;
            const float v = c[j][r];
            binput[off] = v;
            gmsg[off]   = fmaxf(v, 0.0f);
        }
    }
}

// ---------------------------------------------------------------------------
// Neighbor gather-sum: out[row] = sum_j table[graph[row][j]]
// ---------------------------------------------------------------------------
__global__ __launch_bounds__(256) void k_gather(const int*   __restrict__ graph,
                                                const float* __restrict__ tree,
                                                const float* __restrict__ gmsg,
                                                float* __restrict__ out,
                                                int nrows)
{
    const int row = blockIdx.x * 4 + (threadIdx.x >> 6);
    if (row >= nrows) return;
    const int c4 = (threadIdx.x & 63) * 4;
    float4 acc = make_float4(0.f, 0.f, 0.f, 0.f);
#pragma unroll
    for (int j = 0; j < NBR; ++j) {
        const int idx = graph[(size_t)row * NBR + j];
        const float* src = (idx < NM) ? (tree + (size_t)idx * H)
                                      : (gmsg + (size_t)(idx - NM) * H);
        const float4 v = *(const float4*)(src + c4);
        acc.x += v.x; acc.y += v.y; acc.z += v.z; acc.w += v.w;
    }
    *(float4*)(out + (size_t)row * H + c4) = acc;
}

// ---------------------------------------------------------------------------
// GEMM2: gmsg = relu(binput + nei @ W_h)   (K = 256)
// ---------------------------------------------------------------------------
__global__ __launch_bounds__(256) void k_gemm_msg(const float* __restrict__ nei,
                                                  const float* __restrict__ Wh,
                                                  const float* __restrict__ binput,
                                                  float* __restrict__ gmsg)
{
    const int lane  = threadIdx.x & 31;
    const int wave  = blockIdx.x * (blockDim.x >> 5) + (threadIdx.x >> 5);
    const int tileM = wave >> 2;
    const int colBase = (wave & 3) * 64;
    if (tileM * 16 >= NB_) return;
    const int hs  = lane >> 4;
    const int l16 = lane & 15;
    const int aRow = tileM * 16 + l16;

    v8f c[4] = {};
#pragma unroll 4
    for (int k = 0; k < H; k += 4) {
        const int ka = k + hs * 2;
        v2f a;
        a.x = nei[(size_t)aRow * H + ka];
        a.y = nei[(size_t)aRow * H + ka + 1];
#pragma unroll
        for (int j = 0; j < 4; ++j) {
            const int col = colBase + j * 16 + l16;
            v2f b;
            b.x = Wh[(size_t)ka * H + col];
            b.y = Wh[(size_t)(ka + 1) * H + col];
            c[j] = __builtin_amdgcn_wmma_f32_16x16x4_f32(false, a, false, b,
                                                         (short)0, c[j], false, false);
        }
    }
#pragma unroll
    for (int j = 0; j < 4; ++j) {
        const int col = colBase + j * 16 + l16;
#pragma unroll
        for (int r = 0; r < 8; ++r) {
            const int row = tileM * 16 + r + hs * 8;
            const size_t off = (size_t)row * H + col;
            gmsg[off] = fmaxf(c[j][r] + binput[off], 0.0f);
        }
    }
}

// ---------------------------------------------------------------------------
// GEMM3: relu(concat(fatoms, anei) @ W_o_w + W_o_b) fused with atomic
// segment-sum into per-molecule sums.  K = 291 zero-padded to 292.
// ---------------------------------------------------------------------------
__device__ __forceinline__ float fetchA3(const float* fatoms, const float* anei,
                                         int row, int k) {
    if (k < AFD)  return fatoms[(size_t)row * AFD + k];
    if (k < KOUT) return anei[(size_t)row * H + (k - AFD)];
    return 0.0f;
}
__device__ __forceinline__ float fetchB3(const float* Wo, int k, int col) {
    return (k < KOUT) ? Wo[(size_t)k * H + col] : 0.0f;
}

__global__ __launch_bounds__(256) void k_gemm_out(const float* __restrict__ fatoms,
                                                  const float* __restrict__ anei,
                                                  const float* __restrict__ Wo,
                                                  const float* __restrict__ Wob,
                                                  const int*   __restrict__ scope,
                                                  float* __restrict__ sums)
{
    const int lane  = threadIdx.x & 31;
    const int wave  = blockIdx.x * (blockDim.x >> 5) + (threadIdx.x >> 5);
    const int tileM = wave >> 2;
    const int colBase = (wave & 3) * 64;
    if (tileM * 16 >= NA) return;
    const int hs  = lane >> 4;
    const int l16 = lane & 15;
    const int aRow = tileM * 16 + l16;

    v8f c[4] = {};
    for (int k = 0; k < 292; k += 4) {
        const int ka = k + hs * 2;
        v2f a;
        a.x = fetchA3(fatoms, anei, aRow, ka);
        a.y = fetchA3(fatoms, anei, aRow, ka + 1);
#pragma unroll
        for (int j = 0; j < 4; ++j) {
            const int col = colBase + j * 16 + l16;
            v2f b;
            b.x = fetchB3(Wo, ka, col);
            b.y = fetchB3(Wo, ka + 1, col);
            c[j] = __builtin_amdgcn_wmma_f32_16x16x4_f32(false, a, false, b,
                                                         (short)0, c[j], false, false);
        }
    }
#pragma unroll
    for (int j = 0; j < 4; ++j) {
        const int col = colBase + j * 16 + l16;
        const float bias = Wob[col];
#pragma unroll
        for (int r = 0; r < 8; ++r) {
            const int row = tileM * 16 + r + hs * 8;
            const float v = fmaxf(c[j][r] + bias, 0.0f);
            const int mol = scope[row];
            atomicAdd(&sums[(size_t)mol * H + col], v);
        }
    }
}

// ---------------------------------------------------------------------------
__global__ __launch_bounds__(256) void k_count(const int* __restrict__ scope,
                                               float* __restrict__ counts, int n)
{
    const int i = blockIdx.x * blockDim.x + threadIdx.x;
    if (i < n) atomicAdd(&counts[scope[i]], 1.0f);
}

__global__ __launch_bounds__(256) void k_finalize(const float* __restrict__ sums,
                                                  const float* __restrict__ counts,
                                                  float* __restrict__ out)
{
    const int i = blockIdx.x * blockDim.x + threadIdx.x;
    if (i < NMOLS * H) out[i] = sums[i] / fmaxf(counts[i >> 8], 1.0f);
}

// ---------------------------------------------------------------------------
extern "C" void kernel_launch(void* const* d_in, const int* in_sizes, int n_in,
                              void* d_out, int out_size, void* d_ws, size_t ws_size,
                              hipStream_t stream)
{
    (void)in_sizes; (void)n_in; (void)out_size; (void)ws_size;
    const float* fatoms = (const float*)d_in[0];
    const float* fbonds = (const float*)d_in[1];
    const int*   agraph = (const int*)d_in[2];
    const int*   bgraph = (const int*)d_in[3];
    const float* tree   = (const float*)d_in[4];
    const int*   scope  = (const int*)d_in[5];
    const float* Wi     = (const float*)d_in[6];
    const float* Wh     = (const float*)d_in[7];
    const float* Wo     = (const float*)d_in[8];
    const float* Wob    = (const float*)d_in[9];
    float* out = (float*)d_out;

    float* ws = (float*)d_ws;
    const size_t BH = (size_t)NB_ * H;
    float* binput = ws;
    float* gmsg   = ws + BH;
    float* nei    = ws + 2 * BH;
    float* sums   = ws + 3 * BH;
    float* counts = sums + (size_t)NMOLS * H;

    hipMemsetAsync(sums, 0, ((size_t)NMOLS * H + NMOLS) * sizeof(float), stream);

    const int wavesPerBlk = 256 / 32;
    const int blkB = (NB_ / 16) * (H / 64) / wavesPerBlk;   // 6250
    const int blkA = (NA  / 16) * (H / 64) / wavesPerBlk;   // 3125

    k_gemm_in<<<blkB, 256, 0, stream>>>(fbonds, Wi, binput, gmsg);
    for (int d = 0; d < 2; ++d) {
        k_gather<<<NB_ / 4, 256, 0, stream>>>(bgraph, tree, gmsg, nei, NB_);
        k_gemm_msg<<<blkB, 256, 0, stream>>>(nei, Wh, binput, gmsg);
    }
    k_gather<<<NA / 4, 256, 0, stream>>>(agraph, tree, gmsg, nei, NA);
    k_gemm_out<<<blkA, 256, 0, stream>>>(fatoms, nei, Wo, Wob, scope, sums);
    k_count<<<(NA + 255) / 256, 256, 0, stream>>>(scope, counts, NA);
    k_finalize<<<(NMOLS * H + 255) / 256, 256, 0, stream>>>(sums, counts, out);
}